// Attention_80247168958499
// MI455X (gfx1250) — compile-verified
//
#include <hip/hip_runtime.h>
#include <hip/hip_bf16.h>

#define Bb 32
#define Ll 4096
#define Dd 512

typedef float v2f __attribute__((ext_vector_type(2)));
typedef float v8f __attribute__((ext_vector_type(8)));

// ---------------------------------------------------------------------------
// Kernel 1: dec_fea[32,512] = s_t_hat[32,512] @ W_dec^T[512,512] + b_dec
// One wave per 16x16 output tile using V_WMMA_F32_16X16X4_F32 (full fp32).
// A tile layout: lane = g*16 + m (g = lane>>4), VGPR v holds K = 2g+v.
// B tile layout: lane = g*16 + n,               VGPR v holds K = 2g+v.
// C/D layout:    VGPR r holds M = r + 8g, N = lane&15.
// ---------------------------------------------------------------------------
__global__ void decproj_wmma_kernel(const float* __restrict__ s_t_hat,
                                    const float* __restrict__ W_dec,
                                    const float* __restrict__ b_dec,
                                    float* __restrict__ dec_fea) {
    const int wave   = (blockIdx.x * blockDim.x + threadIdx.x) >> 5; // 0..63
    const int lane   = threadIdx.x & 31;
    const int m_tile = wave >> 5;   // 0..1  (batch rows)
    const int n_tile = wave & 31;   // 0..31 (output cols)
    const int g      = lane >> 4;
    const int h      = lane & 15;

    const float* arow = s_t_hat + (size_t)(m_tile * 16 + h) * Dd; // A row (M = h)
    const float* brow = W_dec   + (size_t)(n_tile * 16 + h) * Dd; // B^T row (N = h), K contiguous

    v8f c = {0.f, 0.f, 0.f, 0.f, 0.f, 0.f, 0.f, 0.f};

    for (int k0 = 0; k0 < Dd; k0 += 4) {
        const float2 a2 = *(const float2*)(arow + k0 + 2 * g);
        const float2 b2 = *(const float2*)(brow + k0 + 2 * g);
        v2f a; a.x = a2.x; a.y = a2.y;
        v2f b; b.x = b2.x; b.y = b2.y;
        // 8 args: (neg_a, A, neg_b, B, c_mod, C, reuse_a, reuse_b)
        c = __builtin_amdgcn_wmma_f32_16x16x4_f32(
            false, a, false, b, (short)0, c, false, false);
    }

    const int   n    = n_tile * 16 + h;
    const float bias = b_dec[n];
#pragma unroll
    for (int r = 0; r < 8; ++r) {
        const int m = m_tile * 16 + 8 * g + r;
        dec_fea[(size_t)m * Dd + n] = c[r] + bias;
    }
}

// ---------------------------------------------------------------------------
// Kernel 2: wc_dot[b] = dot(w_c, dec_fea[b,:])   (one wave per batch)
// ---------------------------------------------------------------------------
__global__ void wcdot_kernel(const float* __restrict__ dec_fea,
                             const float* __restrict__ w_c,
                             float* __restrict__ wc_dot) {
    const int b    = threadIdx.x >> 5;   // 32 waves == 32 batches
    const int lane = threadIdx.x & 31;
    const float* df = dec_fea + (size_t)b * Dd;
    float acc = 0.f;
#pragma unroll
    for (int j = lane; j < Dd; j += 32) acc += df[j] * w_c[j];
#pragma unroll
    for (int m = 16; m; m >>= 1) acc += __shfl_xor(acc, m, 32);
    if (lane == 0) wc_dot[b] = acc;
}

// ---------------------------------------------------------------------------
// Kernel 3: scores[b,l] = dot(enc_feat[b,l,:], dec_fea[b,:])
//                         + coverage[b,l]*wc_dot[b] + mask[b,l]
// grid (L/64, B), 256 threads (8 waves); dec_fea[b] cached in LDS as float4.
// Each wave handles 8 rows; per row fully-coalesced float4 loads + shuffle
// reduction.
// ---------------------------------------------------------------------------
__global__ void scores_kernel(const float* __restrict__ encoder_feature,
                              const float* __restrict__ coverage,
                              const float* __restrict__ enc_padding_mask,
                              const float* __restrict__ dec_fea,
                              const float* __restrict__ wc_dot,
                              float* __restrict__ scores) {
    const int b   = blockIdx.y;
    const int tid = threadIdx.x;

    __shared__ float4 decf[Dd / 4]; // 128 float4 = 2KB
    if (tid < Dd / 4) decf[tid] = ((const float4*)(dec_fea + (size_t)b * Dd))[tid];
    __syncthreads();

    const int wave = tid >> 5;
    const int lane = tid & 31;
    const float wdot = wc_dot[b];

#pragma unroll
    for (int rr = 0; rr < 8; ++rr) {
        const int l = blockIdx.x * 64 + wave * 8 + rr;
        const float4* ef = (const float4*)(encoder_feature + ((size_t)b * Ll + l) * Dd);
        float acc = 0.f;
#pragma unroll
        for (int r = 0; r < 4; ++r) {
            const float4 e = ef[r * 32 + lane];
            const float4 f = decf[r * 32 + lane];
            acc += e.x * f.x + e.y * f.y + e.z * f.z + e.w * f.w;
        }
#pragma unroll
        for (int m = 16; m; m >>= 1) acc += __shfl_xor(acc, m, 32);
        if (lane == 0) {
            const size_t idx = (size_t)b * Ll + l;
            scores[idx] = acc + coverage[idx] * wdot + enc_padding_mask[idx];
        }
    }
}

// ---------------------------------------------------------------------------
// Kernel 4: softmax over L (mask already folded in) + coverage update.
// One block of 1024 threads per batch; each thread owns 4 elements.
// ---------------------------------------------------------------------------
__global__ void softmax_kernel(const float* __restrict__ scores,
                               const float* __restrict__ coverage,
                               float* __restrict__ attn,
                               float* __restrict__ cov_new) {
    const int b   = blockIdx.x;
    const int tid = threadIdx.x;
    __shared__ float red[1024];

    float s[4];
#pragma unroll
    for (int j = 0; j < 4; ++j) s[j] = scores[(size_t)b * Ll + j * 1024 + tid];

    float lm = fmaxf(fmaxf(s[0], s[1]), fmaxf(s[2], s[3]));
    red[tid] = lm;
    __syncthreads();
    for (int st = 512; st > 0; st >>= 1) {
        if (tid < st) red[tid] = fmaxf(red[tid], red[tid + st]);
        __syncthreads();
    }
    const float M = red[0];
    __syncthreads();

    float e[4];
    float ls = 0.f;
#pragma unroll
    for (int j = 0; j < 4; ++j) { e[j] = __expf(s[j] - M); ls += e[j]; }
    red[tid] = ls;
    __syncthreads();
    for (int st = 512; st > 0; st >>= 1) {
        if (tid < st) red[tid] += red[tid + st];
        __syncthreads();
    }
    const float inv = 1.0f / red[0];

#pragma unroll
    for (int j = 0; j < 4; ++j) {
        const size_t idx = (size_t)b * Ll + j * 1024 + tid;
        const float a = e[j] * inv;
        attn[idx]    = a;
        cov_new[idx] = coverage[idx] + a;
    }
}

// ---------------------------------------------------------------------------
// Kernel 5: c_t[b,d] = sum_l attn[b,l] * encoder_outputs[b,l,d]
// grid (B, D/64), block (64,4): threadIdx.y picks an L-chunk of 1024 rows,
// partials reduced through LDS. Loads are coalesced across d.
// ---------------------------------------------------------------------------
__global__ void ct_kernel(const float* __restrict__ encoder_outputs,
                          const float* __restrict__ attn,
                          float* __restrict__ c_t) {
    const int b = blockIdx.x;
    const int x = threadIdx.x;
    const int y = threadIdx.y;
    const int d = blockIdx.y * 64 + x;

    const float* eo = encoder_outputs + ((size_t)b * Ll + (size_t)y * 1024) * Dd + d;
    const float* at = attn + (size_t)b * Ll + (size_t)y * 1024;

    float acc = 0.f;
#pragma unroll 4
    for (int i = 0; i < 1024; ++i) acc += at[i] * eo[(size_t)i * Dd];

    __shared__ float sm[4][64];
    sm[y][x] = acc;
    __syncthreads();
    if (y == 0)
        c_t[(size_t)b * Dd + d] = sm[0][x] + sm[1][x] + sm[2][x] + sm[3][x];
}

// ---------------------------------------------------------------------------
extern "C" void kernel_launch(void* const* d_in, const int* in_sizes, int n_in,
                              void* d_out, int out_size, void* d_ws, size_t ws_size,
                              hipStream_t stream) {
    const float* s_t_hat          = (const float*)d_in[0];
    const float* encoder_outputs  = (const float*)d_in[1];
    const float* encoder_feature  = (const float*)d_in[2];
    const float* enc_padding_mask = (const float*)d_in[3];
    const float* coverage         = (const float*)d_in[4];
    const float* W_dec            = (const float*)d_in[5];
    const float* b_dec            = (const float*)d_in[6];
    const float* w_c              = (const float*)d_in[7];

    float* out     = (float*)d_out;
    float* c_t     = out;                        // [B*D]
    float* attn    = out + Bb * Dd;              // [B*L]
    float* cov_new = out + Bb * Dd + Bb * Ll;    // [B*L]

    float* ws      = (float*)d_ws;
    float* dec_fea = ws;                         // [B*D]
    float* wc_dot  = ws + Bb * Dd;               // [B]
    float* scores  = ws + Bb * Dd + 32;          // [B*L]

    decproj_wmma_kernel<<<8, 256, 0, stream>>>(s_t_hat, W_dec, b_dec, dec_fea);
    wcdot_kernel<<<1, 1024, 0, stream>>>(dec_fea, w_c, wc_dot);
    scores_kernel<<<dim3(Ll / 64, Bb), 256, 0, stream>>>(
        encoder_feature, coverage, enc_padding_mask, dec_fea, wc_dot, scores);
    softmax_kernel<<<Bb, 1024, 0, stream>>>(scores, coverage, attn, cov_new);
    ct_kernel<<<dim3(Bb, Dd / 64), dim3(64, 4), 0, stream>>>(
        encoder_outputs, attn, c_t);
}